// MultiviewTemporalSpatialFeatureSampler_60189671686814
// MI455X (gfx1250) — compile-verified
//
#include <hip/hip_runtime.h>
#include <stdint.h>

// ---------------------------------------------------------------------------
// MultiviewTemporalSpatialFeatureSampler for MI455X (gfx1250)
//
// Pure gather workload -> no WMMA. CDNA5-specific path used: Tensor Data
// Mover (tensor_load_to_lds) gathers each point's strided 256-channel column
// (stride = H*W*4 = 45KB) into LDS with ONE descriptor per camera, synced by
// s_wait_tensorcnt, then coalesced b128 LDS->global nontemporal stores.
//
// Toolchain note: this compiler expects the 6-argument builtin form
//   (uint32x4 g0, int32x8 g1, int32x4 g2, int32x4 g3, int32x8 g4, i32 cpol)
// (confirmed by the round-1 diagnostic "expected 6, have 5"). g4 carries no
// information for a plain 2D non-gather, non-iterate descriptor -> zeros.
// ---------------------------------------------------------------------------

#ifndef __has_builtin
#define __has_builtin(x) 0
#endif
#if __has_builtin(__builtin_amdgcn_tensor_load_to_lds)
#define USE_TDM 1
#else
#define USE_TDM 0
#endif

typedef unsigned int u32x4 __attribute__((ext_vector_type(4)));
typedef int          i32x8 __attribute__((ext_vector_type(8)));
typedef int          i32x4 __attribute__((ext_vector_type(4)));
typedef float        f32x4 __attribute__((ext_vector_type(4)));

// Problem constants (from reference setup_inputs)
constexpr int B = 2, T = 4, C = 256, H = 64, W = 176;
constexpr int N = 900, P = 8, NTF = 3, NSF = 3;
constexpr int HW     = H * W;        // 11264
constexpr int THW    = NTF * H * W;  // 33792
constexpr int NP     = N * P;        // 7200
constexpr int POINTS = B * NP;       // 14400
constexpr int OUTC   = 2 * C;        // 512
constexpr int SAMPLED_ELEMS = POINTS * OUTC;  // 7,372,800

#if USE_TDM
// Build a D# for a 2D "column" tile: tile_dim0 = 1 element (4B),
// tile_dim1 = C rows, row stride = H*W elements. TDM packs the C gathered
// elements contiguously into LDS at lds_addr.
__device__ __forceinline__ void tdm_col_load(uint64_t gaddr_bytes, unsigned lds_addr)
{
    u32x4 g0;
    g0.x = 1u;                                        // count=1 (valid), user mode
    g0.y = lds_addr;                                  // LDS byte address
    g0.z = (unsigned)(gaddr_bytes & 0xFFFFFFFFull);   // global_addr[31:0]
    g0.w = ((unsigned)(gaddr_bytes >> 32) & 0x01FFFFFFu) | (2u << 30); // [56:32] | type=2

    i32x8 g1;
    g1[0] = (int)(2u << 16);            // workgroup_mask=0, data_size=2 (4 bytes)
    g1[1] = (int)(1u << 16);            // tensor_dim0 = 1  (bits 63:48)
    g1[2] = (int)((unsigned)C << 16);   // tensor_dim1 = 256 (bits 95:80); dim0 hi = 0
    g1[3] = (int)(1u << 16);            // tile_dim0 = 1 (bits 127:112); dim1 hi = 0
    g1[4] = C;                          // tile_dim1 = 256; tile_dim2 = 0 (2D tile)
    g1[5] = HW;                         // tensor_dim0_stride = 11264 elements
    g1[6] = 0;                          // dim0_stride hi / dim1_stride lo
    g1[7] = 0;                          // dim1_stride hi (unused for 2D)

    i32x4 g2 = {1, 0, 0, 0};            // tensor_dim2=1 (unused), tile_dim3=0
    i32x4 g3 = {0, 0, 0, 0};

    i32x8 g4 = {0, 0, 0, 0, 0, 0, 0, 0}; // extra group (clang-23 form): unused here

    __builtin_amdgcn_tensor_load_to_lds(g0, g1, g2, g3, g4, 0 /*cpol*/);
}
#endif

__global__ void __launch_bounds__(256)
sampler_kernel(const float* __restrict__ cam0, const float* __restrict__ cam1,
               const int* __restrict__ tidx, const int* __restrict__ sidx,
               const int* __restrict__ pidx, float* __restrict__ out)
{
    const int lane = (int)(threadIdx.x & 31u);
    // wave-uniform values go through readfirstlane so the descriptor lands in SGPRs
    const int wave  = __builtin_amdgcn_readfirstlane((int)(threadIdx.x >> 5));
    const int point = (int)blockIdx.x * 8 + wave;      // one point per wave, [0,14400)

    const int b = point / NP;
    const int r = point - b * NP;
    const int n = r >> 3;                               // P == 8
    // p = r & 7 is implicit in `point` ordering

    // fused index chase: p -> (s,h,w) -> sp -> frame
    const int pi  = __builtin_amdgcn_readfirstlane(pidx[point]);
    const int s   = pi / THW;                           // s index (t' is irrelevant)
    const int rem = pi % HW;
    const int h   = rem / W;
    const int w   = rem - h * W;
    const int sp    = __builtin_amdgcn_readfirstlane(sidx[(b * N + n) * NSF + s]);
    const int frame = __builtin_amdgcn_readfirstlane(tidx[(b * N + n) * NTF + sp]);

    const uint64_t elem_off = (uint64_t)(b * T + frame) * (uint64_t)(C * HW)
                            + (uint64_t)(h * W + w);
    float* const ob = out + (size_t)point * OUTC;

#if USE_TDM
    __shared__ __attribute__((aligned(16))) float smem[8][OUTC];
    const unsigned lds0 =
        __builtin_amdgcn_readfirstlane((int)(unsigned)(uintptr_t)(&smem[wave][0]));

    // Two DMA descriptors gather both cameras' 256-channel columns into LDS.
    tdm_col_load((uint64_t)cam0 + (elem_off << 2), lds0);
    tdm_col_load((uint64_t)cam1 + (elem_off << 2), lds0 + (unsigned)(C * 4));
    __builtin_amdgcn_s_wait_tensorcnt(0);

    // 2KB per point: 4 iterations of coalesced b128 LDS reads + NT global stores
#pragma unroll
    for (int it = 0; it < 4; ++it) {
        f32x4 v = *(const f32x4*)(&smem[wave][it * 128 + lane * 4]);
        __builtin_nontemporal_store(v, (f32x4*)(ob + it * 128 + lane * 4));
    }
#else
    // Fallback: direct strided gather, nontemporal (no reuse of feature lines)
    const float* c0 = cam0 + elem_off;
    const float* c1 = cam1 + elem_off;
#pragma unroll
    for (int k = 0; k < 8; ++k) {
        const int c = k * 32 + lane;
        float v0 = __builtin_nontemporal_load(c0 + (size_t)c * HW);
        float v1 = __builtin_nontemporal_load(c1 + (size_t)c * HW);
        __builtin_nontemporal_store(v0, ob + c);
        __builtin_nontemporal_store(v1, ob + C + c);
    }
#endif
}

// Pass-through tail of d_out: temporal (5400), spatial (5400), point (14400)
// indices converted to the output buffer's f32 dtype.
__global__ void passthru_kernel(const int* __restrict__ tidx, const int* __restrict__ sidx,
                                const int* __restrict__ pidx, float* __restrict__ out)
{
    const int i  = (int)(blockIdx.x * blockDim.x + threadIdx.x);
    const int nT = B * N * NTF;   // 5400
    const int nS = B * N * NSF;   // 5400
    const int nP = B * N * P;     // 14400
    if (i < nT)                 out[i] = (float)tidx[i];
    else if (i < nT + nS)       out[i] = (float)sidx[i - nT];
    else if (i < nT + nS + nP)  out[i] = (float)pidx[i - nT - nS];
}

extern "C" void kernel_launch(void* const* d_in, const int* in_sizes, int n_in,
                              void* d_out, int out_size, void* d_ws, size_t ws_size,
                              hipStream_t stream)
{
    (void)in_sizes; (void)n_in; (void)out_size; (void)d_ws; (void)ws_size;

    const float* cam0 = (const float*)d_in[0];
    const float* cam1 = (const float*)d_in[1];
    // d_in[2] = anchor_centers (unused by the sampled output)
    const int* tidx = (const int*)d_in[3];
    const int* sidx = (const int*)d_in[4];
    const int* pidx = (const int*)d_in[5];
    float* out = (float*)d_out;

    // 14400 points, 1 point per wave, 8 waves (256 threads) per block
    sampler_kernel<<<POINTS / 8, 256, 0, stream>>>(cam0, cam1, tidx, sidx, pidx, out);

    const int tail = B * N * (NTF + NSF + P);  // 25200
    passthru_kernel<<<(tail + 255) / 256, 256, 0, stream>>>(tidx, sidx, pidx,
                                                            out + SAMPLED_ELEMS);
}